// RossiDirGraphSAGE_83408264888594
// MI455X (gfx1250) — compile-verified
//
#include <hip/hip_runtime.h>
#include <hip/hip_bf16.h>

typedef __attribute__((ext_vector_type(2))) float v2f;
typedef __attribute__((ext_vector_type(8))) float v8f;

#define F 128
#define KTOT 384
#define KPAD (KTOT + 4)   // LDS row stride in floats: 388 % 64 = 4 -> conflict-free, 8B aligned

// ---------------- degree accumulation (once; graph is static) ----------------
__global__ void deg_kernel(const int* __restrict__ src, const int* __restrict__ dst,
                           float* __restrict__ degi, float* __restrict__ dego, int n_edges) {
    int e = blockIdx.x * blockDim.x + threadIdx.x;
    if (e < n_edges) {
        unsafeAtomicAdd(&degi[dst[e]], 1.0f);
        unsafeAtomicAdd(&dego[src[e]], 1.0f);
    }
}

// ---------------- edge scatter: wave-per-edge, full 512B row per wave ----------------
__global__ void scatter_kernel(const float* __restrict__ h, const int* __restrict__ src,
                               const int* __restrict__ dst, float* __restrict__ accf,
                               float* __restrict__ accb, int n_edges) {
    int gid  = blockIdx.x * blockDim.x + threadIdx.x;
    int edge = gid >> 5;          // wave32
    int lane = gid & 31;
    if (edge >= n_edges) return;
    int s = src[edge];
    int d = dst[edge];
    const float4* hs = (const float4*)(h + (size_t)s * F);
    const float4* hd = (const float4*)(h + (size_t)d * F);
    float4 vs = hs[lane];         // 32 lanes x 16B = one full row
    float4 vd = hd[lane];
    float* af = accf + (size_t)d * F + lane * 4;
    float* ab = accb + (size_t)s * F + lane * 4;
    unsafeAtomicAdd(af + 0, vs.x); unsafeAtomicAdd(af + 1, vs.y);
    unsafeAtomicAdd(af + 2, vs.z); unsafeAtomicAdd(af + 3, vs.w);
    unsafeAtomicAdd(ab + 0, vd.x); unsafeAtomicAdd(ab + 1, vd.y);
    unsafeAtomicAdd(ab + 2, vd.z); unsafeAtomicAdd(ab + 3, vd.w);
}

// ---------------- fused mean + concat-GEMM + bias (+relu), fp32 WMMA ----------------
// out[m, n] = act( sum_k A[m,k] * W[k,n] + b[n] ),  A = [h | accf/deg_in | accb/deg_out]
// Tile: 16 rows x 128 cols per block; 4 waves, each owning 2 column tiles of 16.
// N_NODES = 100000 is an exact multiple of 16, so no partial tiles.
__global__ __launch_bounds__(128) void gemm_kernel(
    const float* __restrict__ h, const float* __restrict__ accf, const float* __restrict__ accb,
    const float* __restrict__ degi, const float* __restrict__ dego,
    const float* __restrict__ W, const float* __restrict__ bias,
    float* __restrict__ out, int relu) {
    __shared__ float Atile[16][KPAD];   // ~24.25 KB

    const int tid  = threadIdx.x;
    const int row0 = blockIdx.x * 16;

    // Stage scaled/concatenated A tile into LDS.
    for (int idx = tid; idx < 16 * KTOT; idx += 128) {
        int r = idx / KTOT;
        int c = idx - r * KTOT;
        int rg = row0 + r;
        float v;
        if (c < F) {
            v = h[(size_t)rg * F + c];
        } else if (c < 2 * F) {
            v = accf[(size_t)rg * F + (c - F)] * (1.0f / fmaxf(degi[rg], 1.0f));
        } else {
            v = accb[(size_t)rg * F + (c - 2 * F)] * (1.0f / fmaxf(dego[rg], 1.0f));
        }
        Atile[r][c] = v;
    }
    __syncthreads();

    const int wave = tid >> 5;      // 0..3
    const int lane = tid & 31;
    const int lrow = lane & 15;     // M (for A) / N (for B) within tile
    const int lhi  = lane >> 4;     // selects K pair {0,1} vs {2,3}
    const int n0   = wave * 32;     // this wave's first output column

    v8f c0 = {};
    v8f c1 = {};

    for (int k0 = 0; k0 < KTOT; k0 += 4) {
        const int ka = k0 + 2 * lhi;
        // A fragment 16x4 f32: lane l<16 -> A[l, k0..k0+1]; lane l>=16 -> A[l-16, k0+2..k0+3]
        v2f a;
        a.x = Atile[lrow][ka];
        a.y = Atile[lrow][ka + 1];
        // B fragment 4x16 f32 (mirrored striping): lane -> W[ka, n], W[ka+1, n]
        const float* wp = W + (size_t)ka * F + n0 + lrow;
        v2f b0; b0.x = wp[0];  b0.y = wp[F];
        v2f b1; b1.x = wp[16]; b1.y = wp[F + 16];
        c0 = __builtin_amdgcn_wmma_f32_16x16x4_f32(false, a, false, b0, (short)0, c0, false, false);
        c1 = __builtin_amdgcn_wmma_f32_16x16x4_f32(false, a, false, b1, (short)0, c1, false, false);
    }

    // C/D layout: lane l, VGPR j -> C[M = j + 8*(l>=16)][N = (l&15) + 16*ntile]
    const int ncol = n0 + lrow;
    const float bi0 = bias[ncol];
    const float bi1 = bias[ncol + 16];
#pragma unroll
    for (int j = 0; j < 8; ++j) {
        const int m = row0 + j + 8 * lhi;
        float v0 = c0[j] + bi0;
        float v1 = c1[j] + bi1;
        if (relu) { v0 = fmaxf(v0, 0.0f); v1 = fmaxf(v1, 0.0f); }
        out[(size_t)m * F + ncol]      = v0;
        out[(size_t)m * F + ncol + 16] = v1;
    }
}

extern "C" void kernel_launch(void* const* d_in, const int* in_sizes, int n_in,
                              void* d_out, int out_size, void* d_ws, size_t ws_size,
                              hipStream_t stream) {
    const float* feat = (const float*)d_in[0];
    const int*   src  = (const int*)d_in[1];
    const int*   dst  = (const int*)d_in[2];
    const float* W1   = (const float*)d_in[3];
    const float* b1   = (const float*)d_in[4];
    const float* W2   = (const float*)d_in[5];
    const float* b2   = (const float*)d_in[6];
    float*       out  = (float*)d_out;

    const int n_nodes = in_sizes[0] / F;   // 100000
    const int n_edges = in_sizes[1];       // 800000

    // Workspace layout (floats): degi[N] | dego[N] | accf[N*F] | accb[N*F] | h1[N*F]
    float* degi = (float*)d_ws;
    float* dego = degi + n_nodes;
    float* accf = dego + n_nodes;
    float* accb = accf + (size_t)n_nodes * F;
    float* h1   = accb + (size_t)n_nodes * F;

    const int degBlocks   = (n_edges + 255) / 256;
    const int scatBlocks  = (n_edges * 32 + 255) / 256;   // wave per edge, 8 edges/block
    const int gemmBlocks  = (n_nodes + 15) / 16;

    // Degrees (static across layers)
    hipMemsetAsync(degi, 0, (size_t)2 * n_nodes * sizeof(float), stream);
    deg_kernel<<<degBlocks, 256, 0, stream>>>(src, dst, degi, dego, n_edges);

    // ---- Layer 1: relu ----
    hipMemsetAsync(accf, 0, (size_t)2 * n_nodes * F * sizeof(float), stream);
    scatter_kernel<<<scatBlocks, 256, 0, stream>>>(feat, src, dst, accf, accb, n_edges);
    gemm_kernel<<<gemmBlocks, 128, 0, stream>>>(feat, accf, accb, degi, dego, W1, b1, h1, 1);

    // ---- Layer 2: identity ----
    hipMemsetAsync(accf, 0, (size_t)2 * n_nodes * F * sizeof(float), stream);
    scatter_kernel<<<scatBlocks, 256, 0, stream>>>(h1, src, dst, accf, accb, n_edges);
    gemm_kernel<<<gemmBlocks, 128, 0, stream>>>(h1, accf, accb, degi, dego, W2, b2, out, 0);
}